// MaxUnpoolWithArgmax_9689446219964
// MI455X (gfx1250) — compile-verified
//
#include <hip/hip_runtime.h>

// MaxUnpoolWithArgmax for MI455X (gfx1250).
// Pure scatter: out[argmax[i]] = in[i] over a zero-filled 4x-larger output.
// Zero FLOPs -> bound by HBM at 23.3 TB/s. Traffic: 64 MiB in + 64 MiB idx
// reads, 256 MiB zero-fill + 64 MiB scatter writes ~= 448 MiB -> ~20 us floor.
// All bulk traffic uses non-temporal b128 vector memory ops (single-use data,
// nothing fits usefully in the 192 MB L2). Scatter uses plain stores: argmax
// indices are unique by construction (".add == .set" per the reference), and
// 16.8M scattered f32 atomic RMWs would bottleneck on L2 atomic transaction
// rate well past the bandwidth floor.
//
// Tail handling is done at LAUNCH time: the vector kernels contain no scalar
// fallback loops (keeps the hot kernels to address-gen + VMEM only); a scalar
// tail kernel is launched only when sizes aren't multiples of the vector
// width (never, for the harness's power-of-two shapes).

typedef float v4f __attribute__((ext_vector_type(4)));
typedef int   v4i __attribute__((ext_vector_type(4)));

// ---- Zero fill: 4 x b128 NT stores = 64 B per thread, 2 KiB per wave32. ----
__global__ __launch_bounds__(256) void unpool_zero_vec(v4f* __restrict__ out4,
                                                       int n16) {
    // n16 = number of 4x-float4 (64 B) chunks.
    int t = blockIdx.x * blockDim.x + threadIdx.x;
    if (t < n16) {
        v4f z = (v4f)0.0f;
        v4f* p = out4 + (size_t)t * 4;
        __builtin_nontemporal_store(z, p + 0);  // s_clause of 4x
        __builtin_nontemporal_store(z, p + 1);  // global_store_b128 th:NT
        __builtin_nontemporal_store(z, p + 2);
        __builtin_nontemporal_store(z, p + 3);
    }
}

__global__ __launch_bounds__(256) void unpool_zero_tail(float* __restrict__ out,
                                                        int start, int n) {
    int i = start + blockIdx.x * blockDim.x + threadIdx.x;
    if (i < n) out[i] = 0.0f;
}

// ---- Scatter: float4 + int4 NT loads, 4 scattered b32 stores. ----
// argmax already encodes the global flat output index (batch offset included),
// so the target is simply out[idx[i]]. Index order (b,h,w,c) with C=64 means
// 16 consecutive lanes (4 channels each) cover one pixel's 64 floats -> full-line coalescing.
__global__ __launch_bounds__(256) void unpool_scatter_vec(const float* __restrict__ in,
                                                          const int* __restrict__ idx,
                                                          float* __restrict__ out,
                                                          int n4) {
    // n4 = number of float4 quartets.
    int t = blockIdx.x * blockDim.x + threadIdx.x;
    if (t < n4) {
        int i = t * 4;
        v4f v = __builtin_nontemporal_load((const v4f*)(in + i));   // b128 th:NT
        v4i k = __builtin_nontemporal_load((const v4i*)(idx + i));  // b128 th:NT
        out[k.x] = v.x;
        out[k.y] = v.y;
        out[k.z] = v.z;
        out[k.w] = v.w;
    }
}

__global__ __launch_bounds__(256) void unpool_scatter_tail(const float* __restrict__ in,
                                                           const int* __restrict__ idx,
                                                           float* __restrict__ out,
                                                           int start, int n) {
    int i = start + blockIdx.x * blockDim.x + threadIdx.x;
    if (i < n) out[idx[i]] = in[i];
}

extern "C" void kernel_launch(void* const* d_in, const int* in_sizes, int n_in,
                              void* d_out, int out_size, void* d_ws, size_t ws_size,
                              hipStream_t stream) {
    (void)n_in; (void)d_ws; (void)ws_size;

    const float* in  = (const float*)d_in[0];  // inputs, fp32, B*H*W*C
    const int*   idx = (const int*)d_in[1];    // argmax, int32 (JAX x64 off)
    float*       out = (float*)d_out;          // fp32, B*2H*2W*C

    const int n = in_sizes[0];      // 16,777,216 input elements
    const int m = out_size;         // 67,108,864 output elements

    const int threads = 256;        // 8 wave32 per block

    // --- Zero fill: bulk in 64 B chunks, launch-time tail. ---
    int n16      = m / 16;                        // 64 B chunks
    int zero_rem = m - n16 * 16;
    if (n16 > 0) {
        int blocks = (n16 + threads - 1) / threads;
        unpool_zero_vec<<<blocks, threads, 0, stream>>>((v4f*)out, n16);
    }
    if (zero_rem > 0) {
        int blocks = (zero_rem + threads - 1) / threads;
        unpool_zero_tail<<<blocks, threads, 0, stream>>>(out, n16 * 16, m);
    }

    // --- Scatter: bulk quartets, launch-time tail. Stream-ordered after zero. ---
    int n4       = n / 4;
    int scat_rem = n - n4 * 4;
    if (n4 > 0) {
        int blocks = (n4 + threads - 1) / threads;
        unpool_scatter_vec<<<blocks, threads, 0, stream>>>(in, idx, out, n4);
    }
    if (scat_rem > 0) {
        int blocks = (scat_rem + threads - 1) / threads;
        unpool_scatter_tail<<<blocks, threads, 0, stream>>>(in, idx, out, n4 * 4, n);
    }
}